// Involution_45045617000588
// MI455X (gfx1250) — compile-verified
//
#include <hip/hip_runtime.h>
#include <hip/hip_bf16.h>

// ---------------------------------------------------------------------------
// Involution (B=8, C=256, H=W=56, K=7, GC=16, G=16, RED=64) for gfx1250.
// conv1x1 GEMMs run on V_WMMA_F32_16X16X4_F32 with MT-tall M-supertiles so
// one (strided, expensive) B fragment feeds MT WMMAs per K-step.
// Involution gather-MAC streams through an LDS-staged halo tile.
// ---------------------------------------------------------------------------

typedef __attribute__((ext_vector_type(2))) float v2f;
typedef __attribute__((ext_vector_type(8))) float v8f;

#define BN_   8
#define CC_   256
#define HH_   56
#define WW_   56
#define HW_   (HH_ * WW_)        // 3136
#define KW_   7
#define GC_   16
#define G_    16
#define RED_  64
#define K2_   49
#define PAD_  3
#define EPS_  1e-5f

// ------------------------- WMMA GEMM + BN + ReLU ---------------------------
// Y[b, m, n] = relu( scale[m] * (sum_k W[m,k] * X[b,k,n] + bias[m]) + shift[m] )
// One wave computes an (MT*16) x 16 tile of [M x 3136] per batch; K step 4.
// B fragment loaded once per K-step, reused by all MT WMMAs.
template <int MT>
__global__ __launch_bounds__(256)
void gemm_bn_relu_wmma(const float* __restrict__ Wt,    // [M, Kdim]
                       const float* __restrict__ X,     // [B, Kdim, HW]
                       float* __restrict__ Y,           // [B, M, HW]
                       const float* __restrict__ bias,  // [M]
                       const float* __restrict__ gamma, // [M]
                       const float* __restrict__ beta,  // [M]
                       const float* __restrict__ mean,  // [M]
                       const float* __restrict__ var,   // [M]
                       int M, int Kdim)
{
    const int lane   = threadIdx.x & 31;
    const int waveId = (blockIdx.x * (blockDim.x >> 5)) + (threadIdx.x >> 5);

    const int mSupers       = M / (16 * MT);
    const int nTiles        = HW_ >> 4;            // 196
    const int tilesPerBatch = mSupers * nTiles;
    const int totalTiles    = tilesPerBatch * BN_;
    if (waveId >= totalTiles) return;              // uniform per wave -> EXEC all-1s

    const int b  = waveId / tilesPerBatch;
    const int t  = waveId - b * tilesPerBatch;
    const int mt = t / nTiles;
    const int nt = t - mt * nTiles;
    const int m0 = mt * (16 * MT);
    const int n0 = nt << 4;

    const int lhalf = lane >> 4;                   // 0: K={0,1}; 1: K={2,3}
    const int lmod  = lane & 15;

    const float* __restrict__ Xb = X + (size_t)b * Kdim * HW_;

    v8f acc[MT];
#pragma unroll
    for (int mi = 0; mi < MT; ++mi) acc[mi] = (v8f){};

    for (int k0 = 0; k0 < Kdim; k0 += 4) {
        // B fragment (shared by all MT subtiles):
        // rows k0 + 2*lhalf + {0,1}, column n0+lmod of Xb
        const float* bcol = Xb + (size_t)(k0 + 2 * lhalf) * HW_ + (n0 + lmod);
        v2f bv;
        bv.x = bcol[0];
        bv.y = bcol[HW_];
#pragma unroll
        for (int mi = 0; mi < MT; ++mi) {
            // A fragment: row (m0 + mi*16 + lmod), cols k0 + 2*lhalf + {0,1}
            const float* arow =
                Wt + (size_t)(m0 + mi * 16 + lmod) * Kdim + k0 + 2 * lhalf;
            v2f a;
            a.x = arow[0];
            a.y = arow[1];
            acc[mi] = __builtin_amdgcn_wmma_f32_16x16x4_f32(
                          /*neg_a=*/false, a, /*neg_b=*/false, bv,
                          /*c_mod=*/(short)0, acc[mi],
                          /*reuse_a=*/false, /*reuse_b=*/false);
        }
    }

    // Epilogue: subtile mi, VGPR r holds M = m0 + mi*16 + r + 8*lhalf,
    // N = n0 + lmod.  Fused BN (inference) + ReLU.
    float* __restrict__ Yb = Y + (size_t)b * M * HW_;
#pragma unroll
    for (int mi = 0; mi < MT; ++mi) {
#pragma unroll
        for (int r = 0; r < 8; ++r) {
            const int   m   = m0 + mi * 16 + r + 8 * lhalf;
            const float inv = rsqrtf(var[m] + EPS_);
            const float sc  = gamma[m] * inv;
            const float sh  = beta[m] - mean[m] * sc + bias[m] * sc;
            float v = acc[mi][r] * sc + sh;
            v = v > 0.0f ? v : 0.0f;
            Yb[(size_t)m * HW_ + (n0 + lmod)] = v;
        }
    }
}

// ------------------------------ Involution ---------------------------------
// out[b, g, c, h, w] = sum_{ki,kj} wgt[b, g, ki*7+kj, h, w] * xpad[b, g, c, h+ki, w+kj]
// Block = (b, g, 14x14 spatial tile). x halo (16ch x 20x20) staged in LDS.
#define TILE_ 14
#define HALO_ (TILE_ + 2 * PAD_)                   // 20

__global__ __launch_bounds__(256)
void involution_mac(const float* __restrict__ X,   // [B, C, H, W]
                    const float* __restrict__ Wg,  // [B, G*K2, H, W]
                    float* __restrict__ Out)       // [B, C, H, W]
{
    __shared__ float xs[GC_][HALO_][HALO_ + 1];    // +1 pad: dodge bank conflicts

    const int tilesPerRow = WW_ / TILE_;           // 4
    const int tilesPerImg = tilesPerRow * (HH_ / TILE_); // 16

    int blk = blockIdx.x;
    const int bg = blk / tilesPerImg;
    const int t  = blk - bg * tilesPerImg;
    const int b  = bg / G_;
    const int g  = bg - b * G_;
    const int th = t / tilesPerRow;
    const int tw = t - th * tilesPerRow;
    const int h0 = th * TILE_;
    const int w0 = tw * TILE_;

    // Cooperative halo load: 16 channels x 20x20, zero-padded at borders.
    const float* __restrict__ Xbg = X + ((size_t)b * CC_ + (size_t)g * GC_) * HW_;
    const int nElems = GC_ * HALO_ * HALO_;        // 6400
    for (int i = threadIdx.x; i < nElems; i += blockDim.x) {
        const int c  = i / (HALO_ * HALO_);
        const int r  = i - c * HALO_ * HALO_;
        const int hh = r / HALO_;
        const int ww = r - hh * HALO_;
        const int gh = h0 + hh - PAD_;
        const int gw = w0 + ww - PAD_;
        float v = 0.0f;
        if (gh >= 0 && gh < HH_ && gw >= 0 && gw < WW_)
            v = Xbg[((size_t)c * HH_ + gh) * WW_ + gw];
        xs[c][hh][ww] = v;
    }
    __syncthreads();

    const int p = threadIdx.x;
    if (p >= TILE_ * TILE_) return;                // 196 compute threads
    const int ph = p / TILE_;
    const int pw = p - ph * TILE_;
    const int h  = h0 + ph;
    const int w  = w0 + pw;

    float acc[GC_];
#pragma unroll
    for (int c = 0; c < GC_; ++c) acc[c] = 0.0f;

    // Weight for this pixel: wgt[b, g, k, h, w]; broadcast over 16 channels.
    const float* __restrict__ Wb =
        Wg + ((size_t)b * G_ + g) * (size_t)K2_ * HW_ + (size_t)h * WW_ + w;
#pragma unroll 7
    for (int k = 0; k < K2_; ++k) {
        const float wk = Wb[(size_t)k * HW_];
        const int ki = k / KW_;
        const int kj = k - ki * KW_;
#pragma unroll
        for (int c = 0; c < GC_; ++c)
            acc[c] = fmaf(wk, xs[c][ph + ki][pw + kj], acc[c]);
    }

    float* __restrict__ Ob =
        Out + ((size_t)b * CC_ + (size_t)g * GC_) * HW_ + (size_t)h * WW_ + w;
#pragma unroll
    for (int c = 0; c < GC_; ++c)
        Ob[(size_t)c * HW_] = acc[c];
}

// ------------------------------- Launcher ----------------------------------
extern "C" void kernel_launch(void* const* d_in, const int* in_sizes, int n_in,
                              void* d_out, int out_size, void* d_ws, size_t ws_size,
                              hipStream_t stream)
{
    const float* x  = (const float*)d_in[0];
    const float* w1 = (const float*)d_in[1];
    const float* b1 = (const float*)d_in[2];
    const float* g1 = (const float*)d_in[3];
    const float* be1= (const float*)d_in[4];
    const float* m1 = (const float*)d_in[5];
    const float* v1 = (const float*)d_in[6];
    const float* w2 = (const float*)d_in[7];
    const float* b2 = (const float*)d_in[8];
    const float* g2 = (const float*)d_in[9];
    const float* be2= (const float*)d_in[10];
    const float* m2 = (const float*)d_in[11];
    const float* v2 = (const float*)d_in[12];
    float* out = (float*)d_out;

    // Workspace carve-up: h1 [8,64,3136] then wgt [8,784,3136]  (~85 MB total)
    float* h1  = (float*)d_ws;
    float* wgt = h1 + (size_t)BN_ * RED_ * HW_;

    const int threads = 256;
    const int wavesPerBlock = threads / 32;

    // conv1: M=64 = 4*16 -> MT=4, 1 supertile; waves = 1*196*8 = 1568
    {
        const int totalWaves = (RED_ / (16 * 4)) * (HW_ / 16) * BN_;
        const int blocks = (totalWaves + wavesPerBlock - 1) / wavesPerBlock;
        gemm_bn_relu_wmma<4><<<blocks, threads, 0, stream>>>(
            w1, x, h1, b1, g1, be1, m1, v1, RED_, CC_);
    }

    // conv2: M=784 = 7*(7*16) -> MT=7, 7 supertiles; waves = 7*196*8 = 10976
    {
        const int totalWaves = ((K2_ * G_) / (16 * 7)) * (HW_ / 16) * BN_;
        const int blocks = (totalWaves + wavesPerBlock - 1) / wavesPerBlock;
        gemm_bn_relu_wmma<7><<<blocks, threads, 0, stream>>>(
            w2, h1, wgt, b2, g2, be2, m2, v2, K2_ * G_, RED_);
    }

    // involution: one block per (b, g, 14x14 tile) -> 8*16*16 = 2048 blocks
    {
        const int blocks = BN_ * G_ * (HH_ / TILE_) * (WW_ / TILE_);
        involution_mac<<<blocks, threads, 0, stream>>>(x, wgt, out);
    }
}